// SGC_17892833755695
// MI455X (gfx1250) — compile-verified
//
#include <hip/hip_runtime.h>
#include <math.h>

// ---------------------------------------------------------------------------
// SGC on gfx1250 (MI455X):
//   deg/norm -> 2x edge-scatter propagation (L2-resident, f32 atomics)
//   -> fp32 WMMA classifier (V_WMMA_F32_16X16X4_F32) + fused log_softmax.
// Workspace layout: [dinv: N f32][y: N*128 f32][z: N*128 f32]  (~103 MB)
// ---------------------------------------------------------------------------

typedef __attribute__((ext_vector_type(2))) float v2f;
typedef __attribute__((ext_vector_type(8))) float v8f;

#define D_FEAT 128
#define N_CLS 40
#define CLS_WAVES 2   // waves (16-node tiles) per classifier block

// ---- degree / norm ---------------------------------------------------------

__global__ void sgc_fill_f32(float* __restrict__ p, float v, int n) {
  int i = blockIdx.x * blockDim.x + threadIdx.x;
  if (i < n) p[i] = v;
}

__global__ void sgc_deg_edges(const int* __restrict__ col, float* __restrict__ deg, int nE) {
  int e = blockIdx.x * blockDim.x + threadIdx.x;
  if (e < nE) {
    (void)__hip_atomic_fetch_add(&deg[col[e]], 1.0f, __ATOMIC_RELAXED,
                                 __HIP_MEMORY_SCOPE_AGENT);
  }
}

__global__ void sgc_dinv(float* __restrict__ deg, int n) {
  int i = blockIdx.x * blockDim.x + threadIdx.x;
  if (i < n) {
    float d = deg[i];
    deg[i] = (d > 0.0f) ? (1.0f / sqrtf(d)) : 0.0f;
  }
}

// ---- propagation -----------------------------------------------------------

// out[v] = 2 * dinv[v]^2 * x[v]   (the two self-loops; also zero-initializes out)
__global__ void sgc_selfloop_init(const float* __restrict__ xin,
                                  const float* __restrict__ dinv,
                                  float* __restrict__ xout, int n_nodes) {
  int t = blockIdx.x * blockDim.x + threadIdx.x;  // n_nodes * 32 threads
  int v = t >> 5;
  int f = t & 31;
  if (v < n_nodes) {
    float di = dinv[v];
    float w = 2.0f * di * di;
    float4 x = reinterpret_cast<const float4*>(xin + (size_t)v * D_FEAT)[f];
    float4 r;
    r.x = w * x.x; r.y = w * x.y; r.z = w * x.z; r.w = w * x.w;
    reinterpret_cast<float4*>(xout + (size_t)v * D_FEAT)[f] = r;
  }
}

// One wave32 per edge: lane l moves float4 l of the 128-float row.
// out[col] += dinv[row]*dinv[col] * x[row]
__global__ void sgc_prop_edges(const int* __restrict__ rowIdx,
                               const int* __restrict__ colIdx,
                               const float* __restrict__ dinv,
                               const float* __restrict__ xin,
                               float* __restrict__ xout, int nE) {
  int lane = threadIdx.x & 31;
  int e = blockIdx.x * (blockDim.x >> 5) + (threadIdx.x >> 5);
  e = __builtin_amdgcn_readfirstlane(e);   // provably wave-uniform -> scalar loads
  if (e >= nE) return;
  int r = rowIdx[e];
  int c = colIdx[e];
  float w = dinv[r] * dinv[c];
  float4 v = reinterpret_cast<const float4*>(xin + (size_t)r * D_FEAT)[lane];
  float* dst = xout + (size_t)c * D_FEAT + lane * 4;
  (void)__hip_atomic_fetch_add(dst + 0, w * v.x, __ATOMIC_RELAXED, __HIP_MEMORY_SCOPE_AGENT);
  (void)__hip_atomic_fetch_add(dst + 1, w * v.y, __ATOMIC_RELAXED, __HIP_MEMORY_SCOPE_AGENT);
  (void)__hip_atomic_fetch_add(dst + 2, w * v.z, __ATOMIC_RELAXED, __HIP_MEMORY_SCOPE_AGENT);
  (void)__hip_atomic_fetch_add(dst + 3, w * v.w, __ATOMIC_RELAXED, __HIP_MEMORY_SCOPE_AGENT);
}

// ---- WMMA classifier + log_softmax ----------------------------------------
// One wave per 16-node tile. 3 column tiles cover 48 classes (40 real + 8 masked).
// FP32 WMMA: A 16x4 (2 f32/lane), B 4x16 (2 f32/lane), C/D 16x16 (8 f32/lane).

__global__ void __launch_bounds__(CLS_WAVES * 32)
sgc_classifier_wmma(const float* __restrict__ X,   // [N,128] propagated features
                    const float* __restrict__ W,   // [40,128]
                    const float* __restrict__ bias,// [40]
                    float* __restrict__ out,       // [N,40] log-probs
                    int n_nodes) {
  __shared__ float smem[CLS_WAVES][16][49];  // 48 cols + pad

  const int lane = threadIdx.x & 31;
  const int wave = threadIdx.x >> 5;
  const int tiles = (n_nodes + 15) >> 4;

  int tile = blockIdx.x * CLS_WAVES + wave;
  const bool tvalid = (tile < tiles);
  if (!tvalid) tile = tiles - 1;            // clamp; all waves reach barrier
  const int row0 = tile << 4;

  const int m    = lane & 15;               // row within tile / column within N-tile
  const int koff = (lane >> 4) << 1;        // 0 or 2 : K sub-slot per ISA layout

  int rload = row0 + m;
  if (rload >= n_nodes) rload = n_nodes - 1;
  const float* xrow = X + (size_t)rload * D_FEAT;

  // B(k,n) = W[n][k].  Column tiles n in {m, 16+m, 32+m}; tile 2 masked past 39.
  const float* w0 = W + (size_t)m * D_FEAT;
  const float* w1 = W + (size_t)(16 + m) * D_FEAT;
  const int n2 = 32 + m;
  const float mask2 = (n2 < N_CLS) ? 1.0f : 0.0f;
  const float* w2 = W + (size_t)((n2 < N_CLS) ? n2 : (N_CLS - 1)) * D_FEAT;

  v8f acc0 = {}, acc1 = {}, acc2 = {};

#pragma unroll 8
  for (int k = 0; k < D_FEAT; k += 4) {
    v2f a, b0, b1, b2;
    a.x = xrow[k + koff];
    a.y = xrow[k + koff + 1];
    b0.x = w0[k + koff];          b0.y = w0[k + koff + 1];
    b1.x = w1[k + koff];          b1.y = w1[k + koff + 1];
    b2.x = mask2 * w2[k + koff];  b2.y = mask2 * w2[k + koff + 1];
    acc0 = __builtin_amdgcn_wmma_f32_16x16x4_f32(false, a, false, b0, (short)0, acc0, false, false);
    acc1 = __builtin_amdgcn_wmma_f32_16x16x4_f32(false, a, false, b1, (short)0, acc1, false, false);
    acc2 = __builtin_amdgcn_wmma_f32_16x16x4_f32(false, a, false, b2, (short)0, acc2, false, false);
  }

  // C/D layout: lane holds column n = lane&15; VGPR j holds row j + 8*(lane>>4).
  const int nb = lane & 15;
  const int mh = (lane >> 4) << 3;
#pragma unroll
  for (int j = 0; j < 8; ++j) {
    smem[wave][mh + j][nb]      = acc0[j];
    smem[wave][mh + j][16 + nb] = acc1[j];
    smem[wave][mh + j][32 + nb] = acc2[j];
  }
  __syncthreads();

  if (lane < 16) {
    const float* srow = &smem[wave][lane][0];
    float mx = -3.402823466e38f;
#pragma unroll
    for (int c = 0; c < N_CLS; ++c) mx = fmaxf(mx, srow[c] + bias[c]);
    float s = 0.0f;
#pragma unroll
    for (int c = 0; c < N_CLS; ++c) s += expf(srow[c] + bias[c] - mx);
    const float lse = mx + logf(s);
    const int grow = row0 + lane;
    if (tvalid && grow < n_nodes) {
      float* o = out + (size_t)grow * N_CLS;
#pragma unroll
      for (int c = 0; c < N_CLS; ++c) o[c] = srow[c] + bias[c] - lse;
    }
  }
}

// ---------------------------------------------------------------------------

extern "C" void kernel_launch(void* const* d_in, const int* in_sizes, int n_in,
                              void* d_out, int out_size, void* d_ws, size_t ws_size,
                              hipStream_t stream) {
  const float* X  = (const float*)d_in[0];   // [N,128]
  const int*   EI = (const int*)d_in[1];     // [2,E]
  const float* W  = (const float*)d_in[2];   // [40,128]
  const float* B  = (const float*)d_in[3];   // [40]
  // d_in[4] = K (fixed at 2 by the reference)

  const int n_nodes = in_sizes[0] / D_FEAT;
  const int n_edges = in_sizes[1] / 2;
  const int* rowIdx = EI;
  const int* colIdx = EI + n_edges;

  char* ws = (char*)d_ws;
  float* dinv = (float*)ws;                                       // N f32 (deg -> dinv)
  size_t off = (((size_t)n_nodes * 4) + 255) & ~(size_t)255;
  float* y = (float*)(ws + off);                                  // N*128 f32
  float* z = (float*)(ws + off + (size_t)n_nodes * D_FEAT * 4);   // N*128 f32
  (void)ws_size;

  const int T = 256;
  // degree (two self-loops baked in) -> dinv in place
  sgc_fill_f32<<<(n_nodes + T - 1) / T, T, 0, stream>>>(dinv, 2.0f, n_nodes);
  sgc_deg_edges<<<(n_edges + T - 1) / T, T, 0, stream>>>(colIdx, dinv, n_edges);
  sgc_dinv<<<(n_nodes + T - 1) / T, T, 0, stream>>>(dinv, n_nodes);

  const int featGrid = ((n_nodes * 32) + T - 1) / T;
  const int edgeGrid = (n_edges + 7) / 8;        // 8 waves/block, 1 wave per edge

  // hop 1: y = P x
  sgc_selfloop_init<<<featGrid, T, 0, stream>>>(X, dinv, y, n_nodes);
  sgc_prop_edges<<<edgeGrid, T, 0, stream>>>(rowIdx, colIdx, dinv, X, y, n_edges);
  // hop 2: z = P y
  sgc_selfloop_init<<<featGrid, T, 0, stream>>>(y, dinv, z, n_nodes);
  sgc_prop_edges<<<edgeGrid, T, 0, stream>>>(rowIdx, colIdx, dinv, y, z, n_edges);

  // classifier + log_softmax (WMMA fp32)
  const int tiles = (n_nodes + 15) / 16;
  const int cgrid = (tiles + CLS_WAVES - 1) / CLS_WAVES;
  sgc_classifier_wmma<<<cgrid, CLS_WAVES * 32, 0, stream>>>(z, W, B, (float*)d_out, n_nodes);
}